// ResidueEmbedding_26396869001943
// MI455X (gfx1250) — compile-verified
//
#include <hip/hip_runtime.h>
#include <math.h>

typedef float v2f __attribute__((ext_vector_type(2)));
typedef float v8f __attribute__((ext_vector_type(8)));

#define MT    32          // tokens per block (two 16-row M tiles; B fragments reused across both)
#define DIN   1195        // input feature width
#define XROW  1196        // padded row stride for x in LDS (1196 % 64 == 44 -> 16 distinct banks)
#define H1S   260         // padded stride for h1 (256 wide)
#define H2S   132         // padded stride for h2/h3 (128 wide)

#define WMMA_F32(a, b, c) \
    __builtin_amdgcn_wmma_f32_16x16x4_f32(false, (a), false, (b), (short)0, (c), false, false)

__global__ __launch_bounds__(256) void residue_embed_fused(
    const int* __restrict__ seq, const float* __restrict__ xyz,
    const float* __restrict__ dih, const int* __restrict__ cidx,
    const float* __restrict__ orient,
    const float* __restrict__ aa_emb, const float* __restrict__ chain_emb,
    const float* __restrict__ W1, const float* __restrict__ b1,
    const float* __restrict__ W2, const float* __restrict__ b2,
    const float* __restrict__ W3, const float* __restrict__ b3,
    const float* __restrict__ W4, const float* __restrict__ b4,
    float* __restrict__ out)
{
    __shared__ float xs[MT * XROW];   // dense per-token feature rows (zero-padded) ~150KB
    __shared__ float hA[MT * H1S];    // h1 (32x256), later reused as h3 (32x128, stride H2S)
    __shared__ float hB[MT * H2S];    // h2 (32x128)

    const int tid = threadIdx.x;
    const int t0  = blockIdx.x * MT;

    // ---------------- phase 0: build x tile in LDS ----------------
    for (int i = tid; i < MT * XROW; i += 256) xs[i] = 0.0f;
    __syncthreads();

    // aa_emb gather (cols 0..127) and chain_emb gather (cols 1067..1194)
    for (int i = tid; i < MT * 128; i += 256) {
        int tt = i >> 7, k = i & 127;
        int t  = t0 + tt;
        xs[tt * XROW + k] = aa_emb[seq[t] * 128 + k];
        int ci = cidx[t];
        xs[tt * XROW + 1067 + k] = (ci != 0) ? chain_emb[ci * 128 + k] : 0.0f;
    }
    // coord features: rotate rel-to-CA coords into local frame, scatter into
    // the one-hot slot (cols 128 + seq*45 .. +45). 32 tokens x 15 atoms.
    for (int i = tid; i < MT * 15; i += 256) {
        int tt = i / 15, a = i % 15;
        int t  = t0 + tt;
        const float* xp = xyz + (size_t)t * 45;
        float rx = xp[a * 3 + 0] - xp[3 + 0];   // CA_IDX = 1
        float ry = xp[a * 3 + 1] - xp[3 + 1];
        float rz = xp[a * 3 + 2] - xp[3 + 2];
        const float* O = orient + (size_t)t * 9; // O[j][i]; local_i = sum_j O[j][i]*r_j
        int base = tt * XROW + 128 + seq[t] * 45 + a * 3;
        xs[base + 0] = O[0] * rx + O[3] * ry + O[6] * rz;
        xs[base + 1] = O[1] * rx + O[4] * ry + O[7] * rz;
        xs[base + 2] = O[2] * rx + O[5] * ry + O[8] * rz;
    }
    // dihedral angular encoding (cols 1028..1066): [x, sin(f*x)x6, cos(f*x)x6] per angle
    if (tid < MT * 3) {
        int tt = tid / 3, d = tid % 3;
        float v = dih[(size_t)(t0 + tt) * 3 + d];
        const float FREQ[6] = {1.0f, 2.0f, 3.0f, 1.0f, 0.5f, 1.0f / 3.0f};
        int base = tt * XROW + 1028 + d * 13;
        xs[base] = v;
        #pragma unroll
        for (int f = 0; f < 6; ++f) {
            xs[base + 1 + f] = sinf(FREQ[f] * v);
            xs[base + 7 + f] = cosf(FREQ[f] * v);
        }
    }
    __syncthreads();

    const int lane = tid & 31;
    const int w    = tid >> 5;       // wave id 0..7
    const int half = lane >> 4;      // 0: K pair {0,1}, 1: K pair {2,3}
    const int col  = lane & 15;      // N (B,C,D) or M (A) index

    // ---------------- layer 1: x(32x1195) @ W1(1195x256), relu -> hA ----------------
    // wave w covers N-tiles {w*32, w*32+16}; both 16-row M tiles share B fragments.
    {
        v8f a00 = {}, a01 = {}, a10 = {}, a11 = {};   // acc[mtile][ntile]
        const int n0 = w * 32;
        const float* W1c0 = W1 + n0 + col;
        const float* W1c1 = W1 + n0 + 16 + col;
        const int ar0 = col * XROW;
        const int ar1 = (16 + col) * XROW;
        int kk = half * 2;
        for (int it = 0; it < 298; ++it, kk += 4) {     // steady state, no clamping
            v2f fa0 = *(const v2f*)&xs[ar0 + kk];
            v2f fa1 = *(const v2f*)&xs[ar1 + kk];
            v2f bA, bB;
            bA.x = W1c0[kk * 256];  bA.y = W1c0[kk * 256 + 256];
            bB.x = W1c1[kk * 256];  bB.y = W1c1[kk * 256 + 256];
            a00 = WMMA_F32(fa0, bA, a00);
            a01 = WMMA_F32(fa0, bB, a01);
            a10 = WMMA_F32(fa1, bA, a10);
            a11 = WMMA_F32(fa1, bB, a11);
        }
        {   // tail: rows 1192..1195; row 1195 clamped (its x is zero-padded in LDS)
            v2f fa0 = *(const v2f*)&xs[ar0 + kk];
            v2f fa1 = *(const v2f*)&xs[ar1 + kk];
            int r0 = (kk     < DIN) ? kk     : DIN - 1;
            int r1 = (kk + 1 < DIN) ? kk + 1 : DIN - 1;
            v2f bA, bB;
            bA.x = W1c0[r0 * 256];  bA.y = W1c0[r1 * 256];
            bB.x = W1c1[r0 * 256];  bB.y = W1c1[r1 * 256];
            a00 = WMMA_F32(fa0, bA, a00);
            a01 = WMMA_F32(fa0, bB, a01);
            a10 = WMMA_F32(fa1, bA, a10);
            a11 = WMMA_F32(fa1, bB, a11);
        }
        float biasA = b1[n0 + col], biasB = b1[n0 + 16 + col];
        #pragma unroll
        for (int r = 0; r < 8; ++r) {
            int m = r + half * 8;
            float v00 = a00[r] + biasA, v01 = a01[r] + biasB;
            float v10 = a10[r] + biasA, v11 = a11[r] + biasB;
            hA[m * H1S + n0 + col]             = v00 > 0.0f ? v00 : 0.0f;
            hA[m * H1S + n0 + 16 + col]        = v01 > 0.0f ? v01 : 0.0f;
            hA[(m + 16) * H1S + n0 + col]      = v10 > 0.0f ? v10 : 0.0f;
            hA[(m + 16) * H1S + n0 + 16 + col] = v11 > 0.0f ? v11 : 0.0f;
        }
    }
    __syncthreads();

    // ---------------- layer 2: h1(32x256) @ W2(256x128), relu -> hB ----------------
    {
        v8f ac0 = {}, ac1 = {};
        const int n0 = w * 16;
        for (int k0 = 0; k0 < 256; k0 += 4) {
            int kk = k0 + half * 2;
            v2f fa0 = *(const v2f*)&hA[col * H1S + kk];
            v2f fa1 = *(const v2f*)&hA[(16 + col) * H1S + kk];
            v2f b;
            b.x = W2[kk * 128 + n0 + col];
            b.y = W2[(kk + 1) * 128 + n0 + col];
            ac0 = WMMA_F32(fa0, b, ac0);
            ac1 = WMMA_F32(fa1, b, ac1);
        }
        float bias = b2[n0 + col];
        #pragma unroll
        for (int r = 0; r < 8; ++r) {
            int m = r + half * 8;
            float v0 = ac0[r] + bias, v1 = ac1[r] + bias;
            hB[m * H2S + n0 + col]        = v0 > 0.0f ? v0 : 0.0f;
            hB[(m + 16) * H2S + n0 + col] = v1 > 0.0f ? v1 : 0.0f;
        }
    }
    __syncthreads();

    // ---------------- layer 3: h2(32x128) @ W3(128x128), relu -> hA (stride H2S) ----------------
    {
        v8f ac0 = {}, ac1 = {};
        const int n0 = w * 16;
        for (int k0 = 0; k0 < 128; k0 += 4) {
            int kk = k0 + half * 2;
            v2f fa0 = *(const v2f*)&hB[col * H2S + kk];
            v2f fa1 = *(const v2f*)&hB[(16 + col) * H2S + kk];
            v2f b;
            b.x = W3[kk * 128 + n0 + col];
            b.y = W3[(kk + 1) * 128 + n0 + col];
            ac0 = WMMA_F32(fa0, b, ac0);
            ac1 = WMMA_F32(fa1, b, ac1);
        }
        float bias = b3[n0 + col];
        #pragma unroll
        for (int r = 0; r < 8; ++r) {
            int m = r + half * 8;
            float v0 = ac0[r] + bias, v1 = ac1[r] + bias;
            hA[m * H2S + n0 + col]        = v0 > 0.0f ? v0 : 0.0f;
            hA[(m + 16) * H2S + n0 + col] = v1 > 0.0f ? v1 : 0.0f;
        }
    }
    __syncthreads();

    // ---------------- layer 4: h3(32x128) @ W4(128x128) + b4 -> out ----------------
    {
        v8f ac0 = {}, ac1 = {};
        const int n0 = w * 16;
        for (int k0 = 0; k0 < 128; k0 += 4) {
            int kk = k0 + half * 2;
            v2f fa0 = *(const v2f*)&hA[col * H2S + kk];
            v2f fa1 = *(const v2f*)&hA[(16 + col) * H2S + kk];
            v2f b;
            b.x = W4[kk * 128 + n0 + col];
            b.y = W4[(kk + 1) * 128 + n0 + col];
            ac0 = WMMA_F32(fa0, b, ac0);
            ac1 = WMMA_F32(fa1, b, ac1);
        }
        float bias = b4[n0 + col];
        #pragma unroll
        for (int r = 0; r < 8; ++r) {
            int m = r + half * 8;
            out[(size_t)(t0 + m) * 128 + n0 + col]      = ac0[r] + bias;
            out[(size_t)(t0 + m + 16) * 128 + n0 + col] = ac1[r] + bias;
        }
    }
}

extern "C" void kernel_launch(void* const* d_in, const int* in_sizes, int n_in,
                              void* d_out, int out_size, void* d_ws, size_t ws_size,
                              hipStream_t stream) {
    (void)n_in; (void)d_ws; (void)ws_size; (void)out_size;
    const int*   seq    = (const int*)  d_in[0];
    const float* xyz    = (const float*)d_in[1];
    const float* dih    = (const float*)d_in[2];
    const int*   cidx   = (const int*)  d_in[3];
    const float* orient = (const float*)d_in[4];
    /* d_in[5] = atom_mask — unused by the reference */
    const float* aa_emb    = (const float*)d_in[6];
    const float* chain_emb = (const float*)d_in[7];
    const float* W1 = (const float*)d_in[8];   const float* b1 = (const float*)d_in[9];
    const float* W2 = (const float*)d_in[10];  const float* b2 = (const float*)d_in[11];
    const float* W3 = (const float*)d_in[12];  const float* b3 = (const float*)d_in[13];
    const float* W4 = (const float*)d_in[14];  const float* b4 = (const float*)d_in[15];
    float* out = (float*)d_out;

    const int tokens = in_sizes[0];          // b * L = 16384
    const int blocks = tokens / MT;          // 512
    residue_embed_fused<<<blocks, 256, 0, stream>>>(
        seq, xyz, dih, cidx, orient, aa_emb, chain_emb,
        W1, b1, W2, b2, W3, b3, W4, b4, out);
}